// F0Collisions_6167573037741
// MI455X (gfx1250) — compile-verified
//
#include <hip/hip_runtime.h>
#include <math.h>
#include <stdint.h>

#ifndef __has_builtin
#define __has_builtin(x) 0
#endif

// Device-pass feature detection (host pass sees none of these -> fallback parse path).
#if defined(__AMDGCN__) && __has_builtin(__builtin_amdgcn_wmma_f32_16x16x4_f32)
#define HAVE_WMMA 1
#else
#define HAVE_WMMA 0
#endif

#if defined(__AMDGCN__) && __has_builtin(__builtin_amdgcn_tensor_load_to_lds) && \
    __has_builtin(__builtin_amdgcn_s_wait_tensorcnt)
#define HAVE_TDM 1
#else
#define HAVE_TDM 0
#endif

typedef float v2f  __attribute__((ext_vector_type(2)));
typedef float v8f  __attribute__((ext_vector_type(8)));
typedef unsigned int u32x4 __attribute__((ext_vector_type(4)));
typedef int   i32x4 __attribute__((ext_vector_type(4)));
typedef int   i32x8 __attribute__((ext_vector_type(8)));

namespace {
constexpr int   kNV           = 512;
constexpr int   kBlock        = 256;                  // 8 x wave32
constexpr int   kRowsPerWave  = 32;                   // one row per lane in solve phase
constexpr int   kWaves        = kBlock / 32;
constexpr int   kRowsPerBlock = kWaves * kRowsPerWave; // 256
constexpr int   kChunkK       = 16;                   // K-tile staged per TDM op
constexpr int   kLdsRow       = kChunkK + 2;          // +2 dword pad (TDM pad fields) -> no bank conflicts
constexpr int   kNChunk       = kNV / kChunkK;        // 32
constexpr float kDV           = 8.0f / 512.0f;        // exact power-of-two grid step
constexpr float kInvSqrt2     = 0.70710678118654752440f;
constexpr float kTwoPi        = 6.28318530717958647692f;
constexpr float kFourPi       = 12.56637061435917295385f;
constexpr float kNuee         = 2.221e-07f;
constexpr float kAtol         = 1e-12f;
constexpr float kRtol         = 1e-08f;

// Weight value for B-matrix column `col` at velocity index k (cols >= 3 are 0).
__device__ __forceinline__ float weight_col(int k, int col) {
  const float v  = (k + 0.5f) * kDV;
  const float v2 = v * v;
  if (col == 0) return v2;
  if (col == 1) return v2 * v2;
  if (col == 2) {
    // Sedge = sum_e sqrt(eps_e)*0.5*(f[e]+f[e+1])*(v_edge_e*DV) folds into
    // we[k] = 0.5*DV/sqrt(2) * ( (k*DV)^2 [k>=1] + ((k+1)*DV)^2 [k<=510] ).
    const float cg  = 0.5f * kDV * kInvSqrt2;
    const float vem = (float)k * kDV;
    const float vep = (float)(k + 1) * kDV;
    const float gm  = (k >= 1)       ? vem * vem : 0.0f;
    const float gp  = (k <= kNV - 2) ? vep * vep : 0.0f;
    return cg * (gm + gp);
  }
  return 0.0f;
}
} // namespace

__global__ __launch_bounds__(kBlock)
void vfp_f0_collisions(const float* __restrict__ f0x,
                       const float* __restrict__ dtp,
                       float* __restrict__ out,     // d' during fwd, x after bwd
                       float* __restrict__ cpws)    // c' stream (NX*NV floats in d_ws)
{
  // Weight table in B-fragment layout: wpair[p][col][j] = weight_col(k=2p+j).
  // Lane `16h+col` loads its (K, K+1) pair with one unconditional ds_load_b64;
  // lanes 0-15 hit banks 0-31, lanes 16-31 banks 32-63 -> conflict-free.
  __shared__ float wpair[kNV / 2][16][2];   // 32 KB
  __shared__ float sums[kRowsPerBlock][4];  // per-row WMMA results

  const int tid      = threadIdx.x;
  const int lane     = tid & 31;
  const int wave     = tid >> 5;
  const int rowBlock = blockIdx.x * kRowsPerBlock;

  for (int idx = tid; idx < (kNV / 2) * 16; idx += kBlock) {
    const int p   = idx >> 4;
    const int col = idx & 15;
    wpair[p][col][0] = weight_col(2 * p,     col);
    wpair[p][col][1] = weight_col(2 * p + 1, col);
  }
  __syncthreads();

#if HAVE_WMMA
  // ---- Linear row reductions as fp32 WMMA: D += A(16x4) x W(4x16) --------
  __shared__ float fstage[kWaves][2][kRowsPerWave * kLdsRow];
  const int waveRow0 = rowBlock + wave * kRowsPerWave;
  v8f acc0 = {}; // rows waveRow0 + 0..15
  v8f acc1 = {}; // rows waveRow0 + 16..31

#if HAVE_TDM
  // TDM: 2-D tile (kChunkK x 32 rows) of f32 from global -> padded LDS.
  auto tdm_issue = [&](int chunk, int buf) {
    const uint64_t gaddr = (uint64_t)(uintptr_t)(f0x + (size_t)waveRow0 * kNV +
                                                 (size_t)chunk * kChunkK);
    const uint32_t laddr = (uint32_t)(uintptr_t)&fstage[wave][buf][0];
    u32x4 g0;
    g0.x = 1u;                                       // count=1 (valid user D#)
    g0.y = laddr;                                    // lds_addr (bytes)
    g0.z = (uint32_t)(gaddr & 0xffffffffull);        // global_addr[31:0]
    g0.w = (uint32_t)((gaddr >> 32) & 0x1ffffffull)  // global_addr[56:32]
           | (2u << 30);                             // type=2 ("image")
    i32x8 g1;
    g1.s0 = (int)((2u << 16)      // data_size = 4 bytes
                  | (1u << 20)    // pad_enable
                  | (3u << 22)    // pad_interval: every 16 dwords
                  | (1u << 25));  // pad_amount: 2 dwords  -> LDS row stride 18
    g1.s1 = (int)(((uint32_t)kChunkK & 0xffffu) << 16);      // tensor_dim0[15:0]
    g1.s2 = (int)(((uint32_t)kRowsPerWave & 0xffffu) << 16); // tensor_dim1[15:0]
    g1.s3 = (int)(((uint32_t)kChunkK) << 16);                // tile_dim0
    g1.s4 = (int)kRowsPerWave;                               // tile_dim1 (tile_dim2=0)
    g1.s5 = (int)kNV;                                        // tensor_dim0_stride lo
    g1.s6 = 0;
    g1.s7 = 0;
    const i32x4 gz4 = {0, 0, 0, 0};                          // 2-D: groups 2/3 unused
    const i32x8 gz8 = {0, 0, 0, 0, 0, 0, 0, 0};              // 6-arg variant extra group
    __builtin_amdgcn_tensor_load_to_lds(g0, g1, gz4, gz4, gz8, 0);
  };
  tdm_issue(0, 0);
#endif

  const int m     = lane & 15;          // A-fragment row
  const int khalf = (lane >> 4) << 1;   // K pair offset (0 or 2)
  const int col   = lane & 15;          // B-fragment column

  for (int c = 0; c < kNChunk; ++c) {
    const int buf = c & 1;
#if HAVE_TDM
    if (c + 1 < kNChunk) {
      tdm_issue(c + 1, buf ^ 1);                 // prefetch next chunk
      __builtin_amdgcn_s_wait_tensorcnt(1);      // chunk c complete (in-order)
    } else {
      __builtin_amdgcn_s_wait_tensorcnt(0);
    }
    asm volatile("" ::: "memory");
#else
    for (int idx = lane; idx < kRowsPerWave * kChunkK; idx += 32) {
      const int r  = idx / kChunkK;
      const int kk = idx % kChunkK;
      fstage[wave][buf][r * kLdsRow + kk] =
          f0x[(size_t)(waveRow0 + r) * kNV + (size_t)c * kChunkK + kk];
    }
    __syncthreads();
#endif
    const float* bufp = &fstage[wave][buf][0];
    for (int s = 0; s < kChunkK / 4; ++s) {
      const int kk = s * 4 + khalf;           // within chunk (always even)
      const int kg = c * kChunkK + kk;        // global k (always even)
      v2f a0, a1;
      a0.x = bufp[m * kLdsRow + kk];          // A 16x4: K0/K1 (lanes<16), K2/K3
      a0.y = bufp[m * kLdsRow + kk + 1];
      a1.x = bufp[(m + 16) * kLdsRow + kk];
      a1.y = bufp[(m + 16) * kLdsRow + kk + 1];
      const v2f bw = *(const v2f*)&wpair[kg >> 1][col][0];  // one ds_load_b64
      acc0 = __builtin_amdgcn_wmma_f32_16x16x4_f32(false, a0, false, bw,
                                                   (short)0, acc0, false, false);
      acc1 = __builtin_amdgcn_wmma_f32_16x16x4_f32(false, a1, false, bw,
                                                   (short)0, acc1, false, false);
    }
  }
  // D layout: VGPR j = row M=j (lanes 0-15) / M=j+8 (lanes 16-31), N = lane&15.
  {
    const int half = lane >> 4;
    if (col < 3) {
      #pragma unroll
      for (int j = 0; j < 8; ++j) {
        sums[wave * kRowsPerWave + half * 8 + j][col]      = acc0[j];
        sums[wave * kRowsPerWave + 16 + half * 8 + j][col] = acc1[j];
      }
    }
  }
#else
  // Scalar fallback (also what the host pass parses): per-thread row reduction.
  {
    const float* frow = f0x + (size_t)(rowBlock + tid) * kNV;
    float s2 = 0.f, s4 = 0.f, se = 0.f;
    for (int k = 0; k < kNV; ++k) {
      const float fv = frow[k];
      s2 = fmaf(fv, wpair[k >> 1][0][k & 1], s2);
      s4 = fmaf(fv, wpair[k >> 1][1][k & 1], s4);
      se = fmaf(fv, wpair[k >> 1][2][k & 1], se);
    }
    sums[tid][0] = s2; sums[tid][1] = s4; sums[tid][2] = se;
  }
#endif
  __syncthreads();

  // ---- Per-row scalar phase (one thread = one row) ------------------------
  const int    row  = rowBlock + tid;
  const float  Sv2  = sums[tid][0];
  const float  Sv4  = sums[tid][1];
  const float  Se   = sums[tid][2];
  const float  T_f  = Sv4 / (3.0f * Sv2);
  float beta = 1.0f / T_f;

  // 3-step fixed point: T_M(beta) via 512-point quadrature of exp(-beta v^2/2).
  for (int it = 0; it < 3; ++it) {
    float s2 = 0.f, s4 = 0.f;
    for (int k = 0; k < kNV; ++k) {
      const float v   = (k + 0.5f) * kDV;
      const float v2  = v * v;
      const float t   = __expf(-0.5f * beta * v2);
      const float tv2 = t * v2;
      s2 += tv2;
      s4  = fmaf(tv2, v2, s4);
    }
    const float T_M = s4 / (3.0f * s2);
    const float bn  = beta * T_M / T_f;
    if (fabsf(bn - beta) > kAtol + kRtol * fabsf(beta)) beta = bn;
  }

  const float dt     = dtp[0];
  const float kdt    = dt * kNuee;
  const float inner  = kTwoPi * Se;
  const float n4pi   = kFourPi * Sv2 * kDV;
  const float Dscale = inner / (beta * n4pi);   // D_e = sqrt(eps_e) * Dscale
  const float invDV  = 1.0f / kDV;

  const float* frow = f0x  + (size_t)row * kNV;
  float*       orow = out  + (size_t)row * kNV;
  float*       wrow = cpws + (size_t)row * kNV;

  // ---- Thomas forward sweep; coefficients rebuilt on the fly --------------
  float prevLoT = 0.f, prevHiT = 0.f;
  float cp = 0.f, dp = 0.f;
  for (int i = 0; i < kNV; ++i) {
    float loT = 0.f, hiT = 0.f, sup = 0.f;
    const float vi  = (i + 0.5f) * kDV;
    const float inv = 1.0f / (vi * vi * kDV);
    if (i < kNV - 1) {
      const float ve  = (float)(i + 1) * kDV;        // v_edge
      const float De  = ve * kInvSqrt2 * Dscale;      // sqrt(eps)=ve/sqrt(2)
      const float w   = ve * kDV / De;
      float delta;
      if (fabsf(w) < 1e-6f) delta = 0.5f;
      else                  delta = 1.0f / w - 1.0f / expm1f(w);
      const float lo  = ve * delta - De * invDV;
      const float hi  = ve * (1.0f - delta) + De * invDV;
      const float w2e = ve * ve;
      loT = w2e * lo;
      hiT = w2e * hi;
      sup = hiT * inv;
    }
    const float diag = (loT - prevHiT) * inv;
    const float sub  = (i > 0) ? (-prevLoT * inv) : 0.f;
    const float a    = -kdt * sub;
    const float b    = 1.0f - kdt * diag;
    const float cc   = -kdt * sup;
    const float d    = frow[i];
    const float den  = b - a * cp;
    cp = cc / den;
    dp = (d - a * dp) / den;
    wrow[i] = cp;     // c' stream -> workspace
    orow[i] = dp;     // d' stream -> output buffer (overwritten in bwd sweep)
    prevLoT = loT;
    prevHiT = hiT;
  }
  // ---- Back substitution (in place in d_out) ------------------------------
  float x = 0.f;
  for (int i = kNV - 1; i >= 0; --i) {
    x = orow[i] - wrow[i] * x;
    orow[i] = x;
  }
}

extern "C" void kernel_launch(void* const* d_in, const int* in_sizes, int n_in,
                              void* d_out, int out_size, void* d_ws, size_t ws_size,
                              hipStream_t stream) {
  // setup_inputs order: nu(1), f0x(NX*NV), dt(1), v(NV). nu is unused by the
  // reference solver; v is recomputed analytically ((k+0.5)*DV, DV = 2^-6).
  const float* f0x = (const float*)d_in[1];
  const float* dtp = (const float*)d_in[2];
  float*       out = (float*)d_out;
  float*       ws  = (float*)d_ws;   // needs NX*NV*4 = 32 MB for the c' stream
  const int nx   = in_sizes[1] / kNV;
  const int grid = nx / kRowsPerBlock;
  vfp_f0_collisions<<<grid, kBlock, 0, stream>>>(f0x, dtp, out, ws);
  (void)n_in; (void)out_size; (void)ws_size;
}